// GatSpatial_53257594470969
// MI455X (gfx1250) — compile-verified
//
#include <hip/hip_runtime.h>
#include <math.h>

typedef __attribute__((ext_vector_type(2))) float v2f;
typedef __attribute__((ext_vector_type(4))) float v4f;
typedef __attribute__((ext_vector_type(8))) float v8f;

#define LRELU_ALPHA 0.2f
#define WMMA_F32(A, B, C) \
  __builtin_amdgcn_wmma_f32_16x16x4_f32(false, (A), false, (B), (short)0, (C), false, false)

__device__ __forceinline__ float wave_max32(float v) {
#pragma unroll
  for (int d = 16; d > 0; d >>= 1) v = fmaxf(v, __shfl_xor(v, d, 32));
  return v;
}
__device__ __forceinline__ float wave_sum32(float v) {
#pragma unroll
  for (int d = 16; d > 0; d >>= 1) v += __shfl_xor(v, d, 32);
  return v;
}

// ---------------------------------------------------------------------------
// Kernel 1: adj_norm = D^-1/2 * minmax(Bp + I) * D^-1/2   (32x32, one block)
// ---------------------------------------------------------------------------
__global__ __launch_bounds__(1024) void adjnorm_kernel(const float* __restrict__ Bp,
                                                       float* __restrict__ adjn_out) {
  __shared__ float rmin[1024];
  __shared__ float rmax[1024];
  __shared__ float sm[32 * 33];
  __shared__ float dis[32];
  const int tid = threadIdx.x;
  const int v = tid >> 5, u = tid & 31;
  const float adj = Bp[v * 32 + u] + ((v == u) ? 1.0f : 0.0f);
  rmin[tid] = adj;
  rmax[tid] = adj;
  __syncthreads();
  for (int s = 512; s > 0; s >>= 1) {
    if (tid < s) {
      rmin[tid] = fminf(rmin[tid], rmin[tid + s]);
      rmax[tid] = fmaxf(rmax[tid], rmax[tid + s]);
    }
    __syncthreads();
  }
  const float mn = rmin[0], mx = rmax[0];
  const float anorm = (adj - mn) / (mx - mn);
  sm[v * 33 + u] = anorm;
  __syncthreads();
  if (tid < 32) {
    float s = 0.f;
#pragma unroll
    for (int j = 0; j < 32; ++j) s += sm[tid * 33 + j];
    dis[tid] = rsqrtf(s);
  }
  __syncthreads();
  adjn_out[v * 32 + u] = dis[v] * anorm * dis[u];
}

// ---------------------------------------------------------------------------
// Kernel 2: fully fused GAT-spatial. One block per (n,h) slice.
//   blockDim = 256 (8 wave32), grid = N*H = 512.
// LDS layout (dynamic, floats):
//   shH  [64][8][32]    @0      16384   h slice (async-staged)
//   e1   [32][65]       @16384   2080
//   e2   [32][65]       @18464   2080
//   att  [32][65]       @20544   2080
//   adjs [32][33]       @22624   1056
//   whpS 8 x [16][36]   @23680   4608   per-wave Whp^T scratch (GEMM1 -> GEMM2)
// ---------------------------------------------------------------------------
__global__ __launch_bounds__(256) void gat_fused_kernel(const float* __restrict__ hin,
                                                        const float* __restrict__ W,
                                                        const float* __restrict__ a,
                                                        const float* __restrict__ adjn,
                                                        float* __restrict__ out) {
  extern __shared__ float smem[];
  float* shH  = smem;                  // 16384
  float* e1   = shH + 16384;           // 32*65
  float* e2   = e1 + 32 * 65;          // 32*65
  float* att  = e2 + 32 * 65;          // 32*65
  float* adjs = att + 32 * 65;         // 32*33

  const int tid  = threadIdx.x;
  const int lane = tid & 31;
  const int wave = tid >> 5;
  const int blk  = blockIdx.x;         // = n*64 + h
  const int n    = blk >> 6;
  const int hh   = blk & 63;

  // ---- Phase 1: async-stage h slice (16384 contiguous floats, 64KB) -------
  // GLOBAL_LOAD_ASYNC_TO_LDS_B128, GVS mode: lds_vaddr, global_voffset, s[base]
  {
    const float* gsrc = hin + (size_t)blk * 16384;
    const unsigned ldsBase = (unsigned)(uintptr_t)shH;
#pragma unroll
    for (int i = 0; i < 16; ++i) {
      const unsigned byteOff = (unsigned)(tid + 256 * i) * 16u;
      asm volatile("global_load_async_to_lds_b128 %0, %1, %2"
                   :
                   : "v"(ldsBase + byteOff), "v"(byteOff), "s"(gsrc)
                   : "memory");
    }
  }
  // adj_norm -> LDS, zero att_sum (regular path, overlaps with async DMA)
  for (int i = tid; i < 1024; i += 256) adjs[(i >> 5) * 33 + (i & 31)] = adjn[i];
  for (int i = tid; i < 32 * 65; i += 256) att[i] = 0.f;

  // Fold W·a1, W·a2 into Wa1[t], Wa2[t] (uniform broadcast loads)
  float Wa1[8], Wa2[8];
#pragma unroll
  for (int t = 0; t < 8; ++t) {
    float s1 = 0.f, s2 = 0.f;
#pragma unroll
    for (int o = 0; o < 16; ++o) {
      const float wv = W[t * 16 + o];
      s1 += wv * a[o];
      s2 += wv * a[16 + o];
    }
    Wa1[t] = s1;
    Wa2[t] = s2;
  }
  asm volatile("s_wait_asynccnt 0" ::: "memory");
  __syncthreads();

  // ---- Phase 2: e1[v][w], e2[v][w] = Σ_t h[w][t][v]·Wa{1,2}[t] ------------
  {
    const int v = tid & 31, wb = tid >> 5;
#pragma unroll
    for (int i = 0; i < 8; ++i) {
      const int w = wb + 8 * i;
      float s1 = 0.f, s2 = 0.f;
#pragma unroll
      for (int t = 0; t < 8; ++t) {
        const float hv = shH[(w * 8 + t) * 32 + v];
        s1 += hv * Wa1[t];
        s2 += hv * Wa2[t];
      }
      e1[v * 65 + w] = s1;
      e2[v * 65 + w] = s2;
    }
  }
  __syncthreads();

  // ---- Phase 3: softmax over w (64) for each (v1,v2); sum over v2 ---------
#pragma unroll
  for (int i = 0; i < 4; ++i) {
    const int v1 = wave + 8 * i;
    const float h0 = e1[v1 * 65 + lane];
    const float h1 = e1[v1 * 65 + lane + 32];
    float acc0 = 0.f, acc1 = 0.f;
    for (int v2 = 0; v2 < 32; ++v2) {
      float f0 = h0 + e2[v2 * 65 + lane];
      float f1 = h1 + e2[v2 * 65 + lane + 32];
      f0 = (f0 > 0.f) ? f0 : LRELU_ALPHA * f0;
      f1 = (f1 > 0.f) ? f1 : LRELU_ALPHA * f1;
      const float m = wave_max32(fmaxf(f0, f1));
      const float x0 = __expf(f0 - m);
      const float x1 = __expf(f1 - m);
      const float inv = 1.0f / wave_sum32(x0 + x1);
      acc0 += x0 * inv;
      acc1 += x1 * inv;
    }
    att[v1 * 65 + lane] = acc0;
    att[v1 * 65 + lane + 32] = acc1;
  }
  __syncthreads();

  // ---- Phase 4: two chained WMMA GEMMs per w ------------------------------
  // GEMM1: Whp(v,o) = (att-scaled h^T)(32x8)·W(8x16): M=v (2 tiles), N=o, K=t
  //        (att[v] folded into A rows: only 2 att loads per w per lane)
  // GEMM2: out(o,u) = Whp^T(16x32)·adjn(32x32): M=o, N=u (2 tiles), K=v (8 ksteps)
  const int half2 = (lane >> 4) << 1;   // 0 | 2  (K sub-rows per lane half)
  const int vhalf = (lane >> 4) << 3;   // 0 | 8  (D-layout M offset)
  const int u16   = lane & 15;

  // GEMM2 B fragments (adj_norm), constant over w
  v2f bfrag[2][8];
#pragma unroll
  for (int nt = 0; nt < 2; ++nt)
#pragma unroll
    for (int k = 0; k < 8; ++k) {
      const int r0 = k * 4 + half2;
      const int u  = nt * 16 + u16;
      bfrag[nt][k].x = adjs[r0 * 33 + u];
      bfrag[nt][k].y = adjs[(r0 + 1) * 33 + u];
    }
  // GEMM1 B fragments (W), constant over w
  v2f b1frag[2];
#pragma unroll
  for (int k = 0; k < 2; ++k) {
    const int t0 = k * 4 + half2;
    b1frag[k].x = W[t0 * 16 + u16];
    b1frag[k].y = W[(t0 + 1) * 16 + u16];
  }

  float* whpS = smem + 23680 + wave * (16 * 36);   // per-wave scratch [o][v], stride 36

  for (int wi = 0; wi < 8; ++wi) {
    const int w = wave + 8 * wi;

    // att scales for this lane's A rows (v = u16 and v = 16+u16); conflict-free
    const float s0 = att[u16 * 65 + w];
    const float s1 = att[(16 + u16) * 65 + w];

    // GEMM1 with att pre-scaled A (D = Whp directly)
    v8f wh0 = {};
    v8f wh1 = {};
#pragma unroll
    for (int k = 0; k < 2; ++k) {
      const int t0 = k * 4 + half2;
      v2f a1, a2;
      a1.x = shH[(w * 8 + t0) * 32 + u16] * s0;
      a1.y = shH[(w * 8 + t0 + 1) * 32 + u16] * s0;
      a2.x = shH[(w * 8 + t0) * 32 + 16 + u16] * s1;
      a2.y = shH[(w * 8 + t0 + 1) * 32 + 16 + u16] * s1;
      wh0 = WMMA_F32(a1, b1frag[k], wh0);
      wh1 = WMMA_F32(a2, b1frag[k], wh1);
    }

    // Spill Whp tiles to per-wave scratch: D is contiguous in v per lane
    // (VGPR r <-> v = vhalf + r), so 4x ds_store_b128, 16B aligned (stride 36).
    float* wp = whpS + u16 * 36;
    *(v4f*)(wp + vhalf)          = __builtin_shufflevector(wh0, wh0, 0, 1, 2, 3);
    *(v4f*)(wp + vhalf + 4)      = __builtin_shufflevector(wh0, wh0, 4, 5, 6, 7);
    *(v4f*)(wp + 16 + vhalf)     = __builtin_shufflevector(wh1, wh1, 0, 1, 2, 3);
    *(v4f*)(wp + 16 + vhalf + 4) = __builtin_shufflevector(wh1, wh1, 4, 5, 6, 7);
    // same-wave LDS ordering is in-order; compiler inserts the single s_wait_dscnt

    // GEMM2: A fragments reloaded as aligned v2f (8B aligned, disjoint bank pairs)
    v8f acc0 = {};
    v8f acc1 = {};
#pragma unroll
    for (int k = 0; k < 8; ++k) {
      const int v0 = k * 4 + half2;
      const v2f af = *(const v2f*)(whpS + u16 * 36 + v0);
      acc0 = WMMA_F32(af, bfrag[0][k], acc0);
      acc1 = WMMA_F32(af, bfrag[1][k], acc1);
    }

    // ELU + store. D layout: VGPR r, lanes0-15 -> o=r, lanes16-31 -> o=r+8; u=lane%16
    const size_t base = ((((size_t)n * 64 + w) * 64 + hh) * 16) * 32;
#pragma unroll
    for (int r = 0; r < 8; ++r) {
      float x0v = acc0[r];
      float x1v = acc1[r];
      x0v = (x0v > 0.f) ? x0v : (__expf(x0v) - 1.0f);
      x1v = (x1v > 0.f) ? x1v : (__expf(x1v) - 1.0f);
      const int o = r + vhalf;
      out[base + (size_t)o * 32 + u16]      = x0v;
      out[base + (size_t)o * 32 + 16 + u16] = x1v;
    }
  }
}

// ---------------------------------------------------------------------------
extern "C" void kernel_launch(void* const* d_in, const int* in_sizes, int n_in,
                              void* d_out, int out_size, void* d_ws, size_t ws_size,
                              hipStream_t stream) {
  const float* h  = (const float*)d_in[0];   // (8,64,64,8,32)
  const float* W  = (const float*)d_in[1];   // (8,16)
  const float* a  = (const float*)d_in[2];   // (32,1)
  const float* Bp = (const float*)d_in[3];   // (32,32)
  float* out  = (float*)d_out;               // (8,64,64,16,32)
  float* adjn = (float*)d_ws;                // 1024 floats scratch

  adjnorm_kernel<<<1, 1024, 0, stream>>>(Bp, adjn);

  const size_t smemFloats = 16384 + 3 * 32 * 65 + 32 * 33 + 8 * 16 * 36;
  gat_fused_kernel<<<512, 256, smemFloats * sizeof(float), stream>>>(h, W, a, adjn, out);
}